// Decode_48988396978319
// MI455X (gfx1250) — compile-verified
//
#include <hip/hip_runtime.h>
#include <stdint.h>
#include <stddef.h>

#ifndef __has_builtin
#define __has_builtin(x) 0
#endif

#define N_CAP   524288u   // 1<<19 sort capacity (>= max candidates ~320K)
#define MAX_OUT 400

typedef unsigned int u32x4 __attribute__((ext_vector_type(4)));
typedef unsigned int u32x8 __attribute__((ext_vector_type(8)));

// ---------------- helpers ----------------

__device__ __forceinline__ unsigned ballot32(bool p) {
#if __has_builtin(__builtin_amdgcn_ballot_w32)
  return __builtin_amdgcn_ballot_w32(p);
#else
  return (unsigned)__ballot((int)p);
#endif
}

__device__ __forceinline__ float4 decode_box(float4 o, float4 a) {
#pragma clang fp contract(off)
  float cx = o.x * a.z + a.x;
  float cy = o.y * a.w + a.y;
  float w  = expf(o.z) * a.z;
  float h  = expf(o.w) * a.w;
  float4 b;
  b.x = cx - 0.5f * w;
  b.y = cy - 0.5f * h;
  b.z = cx + 0.5f * w;
  b.w = cy + 0.5f * h;
  return b;
}

__device__ __forceinline__ bool iou_gt_half(float x1, float y1, float x2, float y2, float ar,
                                            float kx1, float ky1, float kx2, float ky2, float kar) {
#pragma clang fp contract(off)
  float ix1 = fmaxf(x1, kx1), iy1 = fmaxf(y1, ky1);
  float ix2 = fminf(x2, kx2), iy2 = fminf(y2, ky2);
  float inter = fmaxf(ix2 - ix1, 0.0f) * fmaxf(iy2 - iy1, 0.0f);
  float iou = inter / (ar + kar - inter);
  return iou > 0.5f;
}

// ---------------- kernels ----------------

// Zero sort keys, output buffer and the candidate counter (graph replays must be deterministic).
__global__ void nms_init_kernel(unsigned long long* __restrict__ keys,
                                unsigned* __restrict__ count,
                                float* __restrict__ out, int out_n) {
  unsigned i = blockIdx.x * blockDim.x + threadIdx.x;
  if (i < N_CAP) keys[i] = 0ULL;
  if (i < (unsigned)out_n) out[i] = 0.0f;
  if (i == 0) *count = 0u;
}

// Threshold filter: pack (score_bits, ~idx) into 64-bit key. Score > 0.7 always,
// so real keys are nonzero; zeros are sort sentinels (sort is descending).
__global__ void nms_filter_kernel(const float* __restrict__ confs,
                                  const float* __restrict__ thrp,
                                  unsigned long long* __restrict__ keys,
                                  unsigned* __restrict__ count, int n) {
  int i = blockIdx.x * blockDim.x + threadIdx.x;
  if (i >= n) return;
  float conf = confs[2 * i];
  if (conf > thrp[0]) {
    unsigned j = atomicAdd(count, 1u);
    if (j < N_CAP) {
      unsigned long long key =
          ((unsigned long long)__float_as_uint(conf) << 32) |
          (unsigned long long)(0xFFFFFFFFu - (unsigned)i);
      keys[j] = key;
    }
  }
}

// Bitonic sort (descending): global-stride pass, stride j >= 2048.
__global__ void bitonic_global_kernel(unsigned long long* __restrict__ keys,
                                      unsigned k, unsigned j) {
  unsigned i = blockIdx.x * blockDim.x + threadIdx.x;
  unsigned p = i ^ j;
  if (p > i) {
    unsigned long long a = keys[i];
    unsigned long long b = keys[p];
    bool sw = ((i & k) == 0u) ? (a < b) : (a > b);
    if (sw) { keys[i] = b; keys[p] = a; }
  }
}

// Bitonic sort (descending): all strides jstart..1 fused in LDS, 2048-key tile per WGP.
// Tile is fetched with one Tensor-Data-Mover DMA (TENSORcnt) and written back with the
// async LDS->memory engine (ASYNCcnt).
__global__ void __launch_bounds__(1024)
bitonic_local_kernel(unsigned long long* __restrict__ keys, unsigned k, unsigned jstart) {
  __shared__ unsigned long long tile[2048];
  const unsigned t = threadIdx.x;
  const unsigned base = blockIdx.x * 2048u;

  // --- TDM: one descriptor-driven DMA pulls the whole 16KB tile into LDS ---
  if (t == 0) {
    unsigned long long ga = (unsigned long long)(size_t)(keys + base);
    unsigned la = (unsigned)(size_t)&tile[0];       // low 32 bits = LDS byte offset
    u32x4 g0;
    g0[0] = 1u;                                     // count=1 (valid user descriptor)
    g0[1] = la;                                     // lds_addr
    g0[2] = (unsigned)(ga & 0xFFFFFFFFull);         // global_addr[31:0]
    g0[3] = (unsigned)((ga >> 32) & 0x01FFFFFFull)  // global_addr[56:32]
            | 0x80000000u;                          // type=2 ("image")
    u32x8 g1;
    g1[0] = 0x3u << 16;                             // data_size=3 (8 bytes)
    g1[1] = (2048u & 0xFFFFu) << 16;                // tensor_dim0[15:0] (bits 63:48)
    g1[2] = 0u;                                     // tensor_dim0 hi / tensor_dim1 lo
    g1[3] = 2048u << 16;                            // tile_dim0 = 2048 (bits 127:112)
    g1[4] = 0u;                                     // tile_dim1=0, tile_dim2=0 (1-D tile)
    g1[5] = 2048u;                                  // tensor_dim0_stride[31:0]
    g1[6] = 0u;
    g1[7] = 0u;
    u32x4 gz;
    gz[0] = 0u; gz[1] = 0u; gz[2] = 0u; gz[3] = 0u; // groups 2/3: dims unused
    asm volatile("tensor_load_to_lds %0, %1, %2, %3"
                 :: "s"(g0), "s"(g1), "s"(gz), "s"(gz) : "memory");
#if __has_builtin(__builtin_amdgcn_s_wait_tensorcnt)
    __builtin_amdgcn_s_wait_tensorcnt(0);
#else
    asm volatile("s_wait_tensorcnt 0x0" ::: "memory");
#endif
  }
  __syncthreads();

  for (unsigned j = jstart; j >= 1u; j >>= 1) {
    unsigned i = ((t & ~(j - 1u)) << 1) | (t & (j - 1u));  // bit j of i is clear
    unsigned p = i + j;
    unsigned gi = base + i;
    unsigned long long a = tile[i];
    unsigned long long b = tile[p];
    bool sw = ((gi & k) == 0u) ? (a < b) : (a > b);
    if (sw) { tile[i] = b; tile[p] = a; }
    __syncthreads();
  }

  // --- async write-back: 16B per lane straight out of LDS ---
  {
    unsigned lds_src = (unsigned)(size_t)&tile[2u * t];
    const unsigned long long* gdst = keys + base + 2u * t;
    asm volatile("global_store_async_from_lds_b128 %0, %1, off"
                 :: "v"(gdst), "v"(lds_src) : "memory");
    asm volatile("s_wait_asynccnt 0" ::: "memory");
  }
}

// Exact greedy NMS as a sorted scan. One workgroup (32 wave32 waves), kept set in LDS,
// per-wave survivor bitmasks resolve intra-chunk order; serial depth ~= #kept.
// Key chunks are double-buffered in LDS via the async-to-LDS engine.
__global__ void __launch_bounds__(1024)
nms_scan_kernel(const unsigned long long* __restrict__ keys,
                const unsigned* __restrict__ countp,
                const float4* __restrict__ offsets,
                const float4* __restrict__ anchors,
                float* __restrict__ out) {
  __shared__ float kx1[MAX_OUT], ky1[MAX_OUT], kx2[MAX_OUT], ky2[MAX_OUT], kar[MAX_OUT];
  __shared__ unsigned long long skey[2][1024];
  __shared__ unsigned smask[32];
  __shared__ int s_kept;

  const int tid  = threadIdx.x;
  const int wave = tid >> 5;
  const int lane = tid & 31;

  if (tid == 0) s_kept = 0;
  unsigned M = *countp;
  if (M > N_CAP) M = N_CAP;

  // Prime the pipeline: async-fetch chunk 0 into buffer 0 (8B per lane).
  if (M > 0) {
    unsigned lds = (unsigned)(size_t)&skey[0][tid];
    const unsigned long long* g = keys + tid;
    asm volatile("global_load_async_to_lds_b64 %0, %1, off"
                 :: "v"(lds), "v"(g) : "memory");
  }
  __syncthreads();

  for (unsigned base = 0; base < M; base += 1024u) {
    const int cur = (int)((base >> 10) & 1u);

    // Own-wave lanes wrote own-wave slots: per-wave wait is sufficient.
    asm volatile("s_wait_asynccnt 0" ::: "memory");

    // Overlap: kick off the next chunk into the other buffer while we resolve this one.
    if (base + 1024u < M) {
      unsigned lds = (unsigned)(size_t)&skey[cur ^ 1][tid];
      const unsigned long long* g = keys + base + 1024u + tid;
      asm volatile("global_load_async_to_lds_b64 %0, %1, off"
                   :: "v"(lds), "v"(g) : "memory");
      if (base + 2048u < M)
        __builtin_prefetch(&keys[base + 2048u + (unsigned)tid], 0, 0);  // global_prefetch_b8
    }

    unsigned long long key = skey[cur][tid];   // zeros beyond M (sorted sentinels)

    bool  alive = false;
    float score = 0.0f, bx1 = 0, by1 = 0, bx2 = 0, by2 = 0, area = 0;
    int   kc0 = s_kept;  // uniform; stable since last barrier

    if (key != 0ULL) {
      unsigned orig = 0xFFFFFFFFu - (unsigned)(key & 0xFFFFFFFFull);
      score = __uint_as_float((unsigned)(key >> 32));
      float4 b = decode_box(offsets[orig], anchors[orig]);
      bx1 = b.x; by1 = b.y; bx2 = b.z; by2 = b.w;
      {
#pragma clang fp contract(off)
        area = (bx2 - bx1) * (by2 - by1);
      }
      alive = true;
      for (int n = 0; n < kc0; ++n) {
        if (iou_gt_half(bx1, by1, bx2, by2, area, kx1[n], ky1[n], kx2[n], ky2[n], kar[n])) {
          alive = false; break;
        }
      }
    }

    unsigned bal = ballot32(alive);
    __syncthreads();                     // previous smask readers done
    if (lane == 0) smask[wave] = bal;
    __syncthreads();

    // Greedy resolution within the chunk, strictly in sorted order.
    for (;;) {
      int kc = s_kept;                   // uniform
      if (kc >= MAX_OUT) break;
      int pos = -1;
      for (int w = 0; w < 32; ++w) {
        unsigned mm = smask[w];
        if (mm) { pos = (w << 5) + __builtin_ctz(mm); break; }
      }
      if (pos < 0) break;

      if (tid == pos) {                  // append the winner
        kx1[kc] = bx1; ky1[kc] = by1; kx2[kc] = bx2; ky2[kc] = by2; kar[kc] = area;
        out[4 * kc + 0] = bx1;
        out[4 * kc + 1] = by1;
        out[4 * kc + 2] = bx2;
        out[4 * kc + 3] = by2;
        out[4 * MAX_OUT + kc] = score;
      }
      __syncthreads();                   // kept entry visible

      if (alive) {
        if (tid == pos) {
          alive = false;
        } else if (iou_gt_half(bx1, by1, bx2, by2, area,
                               kx1[kc], ky1[kc], kx2[kc], ky2[kc], kar[kc])) {
          alive = false;
        }
      }
      unsigned nb = ballot32(alive);
      __syncthreads();                   // old smask readers done
      if (lane == 0) smask[wave] = nb;
      if (tid == 0) s_kept = kc + 1;
      __syncthreads();
    }

    if (s_kept >= MAX_OUT) break;        // uniform; stable since last barrier
  }
  // Untouched outputs were zeroed by nms_init_kernel (matches invalid slots = 0).
}

// ---------------- launcher ----------------

extern "C" void kernel_launch(void* const* d_in, const int* in_sizes, int n_in,
                              void* d_out, int out_size, void* d_ws, size_t ws_size,
                              hipStream_t stream) {
  const float* confs     = (const float*)d_in[0];
  const float* offsets   = (const float*)d_in[1];
  const float* anchors   = (const float*)d_in[2];
  const float* threshold = (const float*)d_in[3];
  float* out = (float*)d_out;

  const int n = in_sizes[0] / 2;  // N_ANCHORS

  unsigned* count            = (unsigned*)d_ws;
  unsigned long long* keys   = (unsigned long long*)((char*)d_ws + 256);

  nms_init_kernel<<<N_CAP / 256, 256, 0, stream>>>(keys, count, out, out_size);
  nms_filter_kernel<<<(n + 255) / 256, 256, 0, stream>>>(confs, threshold, keys, count, n);

  // In-place bitonic sort of N_CAP 64-bit keys, descending.
  for (unsigned k = 2; k <= N_CAP; k <<= 1) {
    unsigned j = k >> 1;
    for (; j >= 2048u; j >>= 1)
      bitonic_global_kernel<<<N_CAP / 256, 256, 0, stream>>>(keys, k, j);
    bitonic_local_kernel<<<N_CAP / 2048, 1024, 0, stream>>>(keys, k, j);
  }

  nms_scan_kernel<<<1, 1024, 0, stream>>>(keys, count,
                                          (const float4*)offsets,
                                          (const float4*)anchors, out);
}